// EdgeFeatureExtractor_51737176047900
// MI455X (gfx1250) — compile-verified
//
#include <hip/hip_runtime.h>

// ---------------------------------------------------------------------------
// EdgeFeatureExtractor for MI455X (gfx1250, wave32)
//
//   B=16, C=8, HDIM=64, H=W=512, E = C*(C-1) = 56, ROWS = B*E = 896
//
//   Phase 1: valid[b,c] = OR over 512*512 int32 mask  (128 MiB -> ~5.8us @ 23.3TB/s)
//   Phase 2: 896-row MLP [128 ->relu-> 64 -> 64] via V_WMMA_F32_16X16X4_F32
// ---------------------------------------------------------------------------

#define B_SZ   16
#define C_CLS  8
#define HDIM   64
#define E_EDG  56                 // C*(C-1)
#define ROWS   (B_SZ * E_EDG)     // 896
#define MASK_HW (512 * 512)       // elements per (b,c) mask
#define CHUNKS  32                // blocks per node for the mask scan
#define SCAN_T  256               // threads per scan block

typedef float v2f __attribute__((ext_vector_type(2)));
typedef float v8f __attribute__((ext_vector_type(8)));

// ---------------------------------------------------------------------------
// Kernel 0: zero the per-node valid flags in workspace (harness poisons ws).
// ---------------------------------------------------------------------------
__global__ void ef_zero_valid(int* __restrict__ valid) {
  if (threadIdx.x < B_SZ * C_CLS) valid[threadIdx.x] = 0;
}

// ---------------------------------------------------------------------------
// Kernel 1: bandwidth-bound mask scan. valid[node] = any nonzero element.
// Each block covers MASK_HW/CHUNKS = 8192 ints with b128 loads; OR-reduction
// allows a deterministic per-thread early exit once a nonzero is seen.
// ---------------------------------------------------------------------------
__global__ void __launch_bounds__(SCAN_T)
ef_mask_scan(const int* __restrict__ masks, int* __restrict__ valid) {
  const int node  = blockIdx.x / CHUNKS;
  const int chunk = blockIdx.x % CHUNKS;
  const int elemsPerChunk = MASK_HW / CHUNKS;          // 8192 ints = 2048 int4
  const int4* p = (const int4*)(masks + (size_t)node * MASK_HW
                                       + (size_t)chunk * elemsPerChunk);
  int acc = 0;
  // 2048 int4 / 256 threads = 8 int4 per thread, coalesced stride-SCAN_T.
  #pragma unroll
  for (int s = 0; s < 8; ++s) {
    int4 v = p[(size_t)s * SCAN_T + threadIdx.x];
    acc |= v.x | v.y | v.z | v.w;
    if (acc) break;                                    // OR is monotone: safe
  }
  __shared__ int bf;
  if (threadIdx.x == 0) bf = 0;
  __syncthreads();
  if (acc) atomicOr(&bf, 1);                           // cheap LDS atomic
  __syncthreads();
  if (threadIdx.x == 0 && bf) atomicOr(&valid[node], 1);
}

// ---------------------------------------------------------------------------
// Kernel 2: edge MLP with fp32 WMMA.
// Block = 128 threads (4 wave32) handles one 16-row M-tile; wave w owns
// output columns [16w, 16w+16).
//
// Fragment layouts (32-bit operands, wave32, per ISA 7.12.2):
//   A frag (16x4):  lane l, comp c ->  A[l&15][k0 + c + 2*(l>>4)]
//   B frag (4x16):  lane l, comp c ->  B[k0 + c + 2*(l>>4)][l&15]
//   C/D    (16x16): lane l, vgpr r ->  C[r + 8*(l>>4)][l&15]
// ---------------------------------------------------------------------------
__global__ void __launch_bounds__(128)
ef_edge_mlp(const float* __restrict__ feat,   // [16][8][64]
            const float* __restrict__ W1g,    // [128][64]
            const float* __restrict__ b1g,    // [64]
            const float* __restrict__ W2g,    // [64][64]
            const float* __restrict__ b2g,    // [64]
            const int*   __restrict__ valid,  // [128]
            int*         __restrict__ out_idx,  // [16][2][56] int32
            float*       __restrict__ out_attr) // [896][64]  fp32
{
  // Padded strides: A-frag gathers walk rows at fixed column; stride 130/66
  // gives bank index 2*m + k (conflict-free) and keeps 8B alignment (even).
  __shared__ float A_s[16][130];   // x = [feat_i | feat_j], 16 rows x 128
  __shared__ float H_s[16][66];    // relu(x@W1 + b1), 16 rows x 64
  __shared__ float b1_s[64], b2_s[64];
  __shared__ int   rowvalid[16];

  const int tid     = threadIdx.x;
  const int rowBase = blockIdx.x * 16;

  if (tid < 64) { b1_s[tid] = b1g[tid]; b2_s[tid] = b2g[tid]; }

  // Per-row metadata + edge_index writes (ordered-pair decode of np.nonzero(~eye)):
  // e = i*7 + (j<i ? j : j-1)  =>  i = e/7, t = e%7, j = t<i ? t : t+1
  if (tid < 16) {
    const int row = rowBase + tid;
    const int b = row / E_EDG, e = row % E_EDG;
    const int i = e / 7, t = e % 7;
    const int j = (t < i) ? t : t + 1;
    rowvalid[tid] = valid[b * C_CLS + i] & valid[b * C_CLS + j];
    out_idx[b * 2 * E_EDG + e]         = i;
    out_idx[b * 2 * E_EDG + E_EDG + e] = j;
  }

  // Stage A tile: 16 rows x 128 (concat of the two endpoint feature vectors).
  for (int idx = tid; idx < 16 * 128; idx += 128) {
    const int r = idx >> 7, k = idx & 127;
    const int row = rowBase + r;
    const int b = row / E_EDG, e = row % E_EDG;
    const int i = e / 7, t = e % 7;
    const int j = (t < i) ? t : t + 1;
    const int cls = (k < HDIM) ? i : j;
    A_s[r][k] = feat[(b * C_CLS + cls) * HDIM + (k & (HDIM - 1))];
  }
  __syncthreads();

  const int lane = tid & 31;
  const int wave = tid >> 5;
  const int n0   = wave * 16;        // this wave's N-slice
  const int m    = lane & 15;        // A row / B,C column within tile
  const int g    = lane >> 4;        // lane-half selector
  const int nn   = n0 + m;           // global output column for B/C frags

  // ---- Layer 1: H = relu(A[16x128] @ W1[128x64] + b1) ----
  v8f acc = {0.f, 0.f, 0.f, 0.f, 0.f, 0.f, 0.f, 0.f};
  #pragma unroll
  for (int k0 = 0; k0 < 2 * HDIM; k0 += 4) {
    const int k = k0 + 2 * g;
    v2f a = *(const v2f*)&A_s[m][k];              // ds_load_b64, bank-clean
    v2f bfrag;
    bfrag.x = W1g[k * HDIM + nn];                 // weights stay in L2/WGP$
    bfrag.y = W1g[(k + 1) * HDIM + nn];
    acc = __builtin_amdgcn_wmma_f32_16x16x4_f32(
        /*neg_a=*/false, a, /*neg_b=*/false, bfrag,
        /*c_mod=*/(short)0, acc, /*reuse_a=*/false, /*reuse_b=*/false);
  }
  #pragma unroll
  for (int r = 0; r < 8; ++r) {
    const int mo = r + 8 * g;
    const float v = acc[r] + b1_s[nn];
    H_s[mo][nn] = v > 0.f ? v : 0.f;
  }
  __syncthreads();

  // ---- Layer 2: out = H[16x64] @ W2[64x64] + b2, masked by edge validity ----
  v8f acc2 = {0.f, 0.f, 0.f, 0.f, 0.f, 0.f, 0.f, 0.f};
  #pragma unroll
  for (int k0 = 0; k0 < HDIM; k0 += 4) {
    const int k = k0 + 2 * g;
    v2f a = *(const v2f*)&H_s[m][k];
    v2f bfrag;
    bfrag.x = W2g[k * HDIM + nn];
    bfrag.y = W2g[(k + 1) * HDIM + nn];
    acc2 = __builtin_amdgcn_wmma_f32_16x16x4_f32(
        false, a, false, bfrag, (short)0, acc2, false, false);
  }
  #pragma unroll
  for (int r = 0; r < 8; ++r) {
    const int mo  = r + 8 * g;
    const int row = rowBase + mo;
    float v = acc2[r] + b2_s[nn];
    if (!rowvalid[mo]) v = 0.f;
    out_attr[(size_t)row * HDIM + nn] = v;
  }
}

// ---------------------------------------------------------------------------
extern "C" void kernel_launch(void* const* d_in, const int* in_sizes, int n_in,
                              void* d_out, int out_size, void* d_ws, size_t ws_size,
                              hipStream_t stream) {
  const float* feat  = (const float*)d_in[0];  // [16,8,64]
  const int*   masks = (const int*)  d_in[1];  // [16,8,512,512]
  const float* W1    = (const float*)d_in[2];  // [128,64]
  const float* b1    = (const float*)d_in[3];  // [64]
  const float* W2    = (const float*)d_in[4];  // [64,64]
  const float* b2    = (const float*)d_in[5];  // [64]

  int*   valid    = (int*)d_ws;                          // 128 ints of scratch
  int*   out_idx  = (int*)d_out;                         // [16][2][56] int32
  float* out_attr = (float*)d_out + B_SZ * 2 * E_EDG;    // [896][64] fp32

  ef_zero_valid<<<1, 128, 0, stream>>>(valid);
  ef_mask_scan<<<B_SZ * C_CLS * CHUNKS, SCAN_T, 0, stream>>>(masks, valid);
  ef_edge_mlp<<<ROWS / 16, 128, 0, stream>>>(feat, W1, b1, W2, b2, valid,
                                             out_idx, out_attr);
}